// GAT_4836133175806
// MI455X (gfx1250) — compile-verified
//
#include <hip/hip_runtime.h>
#include <hip/hip_bf16.h>

// ---------------------------------------------------------------------------
// 2-layer diagonal-weight GAT for MI455X (gfx1250, wave32).
//
// Cost model: edge aggregation dominates (E=1.7M edges x 128 f32 gather +
// f32 atomic scatter, x2 heads x2 layers). Working set (~150 MB) fits the
// 192 MB L2, so the kernel is L2-bandwidth/atomic bound; FLOPs are ~2 GFLOP.
// Node-level attention coefficients (4 dot products of length 128 per node)
// are computed with V_WMMA_F32_16X16X4_F32, 16 nodes per wave, with the
// B-matrix columns staged in LDS to keep the K-loop branch-free.
// ---------------------------------------------------------------------------

typedef float v2f __attribute__((ext_vector_type(2)));
typedef float v8f __attribute__((ext_vector_type(8)));

#define GAT_F 128

// ---------------------------------------------------------------------------
// Node kernel: for 16 nodes per wave compute, via WMMA f32 16x16x4,
//   st[0][i] = dot(x[i], W0 .* a_src0)   st[1][i] = dot(x[i], W0 .* a_dst0)
//   st[2][i] = dot(x[i], W1 .* a_src1)   st[3][i] = dot(x[i], W1 .* a_dst1)
// A-matrix (16x4 f32): lanes 0-15 hold rows M=0..15 with K=0,1 in the two
// VGPRs; lanes 16-31 hold the same rows with K=2,3 (ISA 7.12.2).
// B-matrix (4x16 f32): column n = lane&15; cols 0..3 carry the (W .* a)
// vectors, cols 4..15 zero.  The 4 real columns + 1 zero column are staged
// into LDS once per block so the K-loop has no divergence.
// ---------------------------------------------------------------------------
__global__ void gat_node_wmma(const float* __restrict__ x,
                              const float* __restrict__ W,   // (2, F)
                              const float* __restrict__ A,   // (2, 2F) = asrc0,adst0,asrc1,adst1
                              float* __restrict__ st,        // (4, N)
                              int N)
{
    __shared__ float bcols[5 * GAT_F];            // cols 0..3 = W.*a, col 4 = 0

    // Stage B columns: 640 entries, 256 threads.
    for (int idx = threadIdx.x; idx < 5 * GAT_F; idx += blockDim.x) {
        const int col = idx >> 7;                  // 0..4
        const int k   = idx & (GAT_F - 1);
        bcols[idx] = (col < 4) ? A[col * GAT_F + k] * W[(col >> 1) * GAT_F + k]
                               : 0.0f;
    }
    __syncthreads();

    const int lane = threadIdx.x & 31;
    const int wave = threadIdx.x >> 5;
    const int base = (blockIdx.x * 8 + wave) * 16;   // 8 waves per 256-thread block
    if (base >= N) return;                            // wave-uniform: EXEC stays full

    const int m    = lane & 15;       // A: row M      B: column N
    const int half = lane >> 4;       // 0 -> K=0,1    1 -> K=2,3
    const int bn   = (m < 4) ? m : 4; // clamp to the zero column

    const float* arow = x + (size_t)(base + m) * GAT_F + half * 2;
    const float* bcol = bcols + bn * GAT_F + half * 2;

    v8f c = {};
    #pragma unroll 8
    for (int k = 0; k < GAT_F; k += 4) {
        v2f a, b;
        a.x = arow[k];
        a.y = arow[k + 1];
        b.x = bcol[k];
        b.y = bcol[k + 1];
        // (neg_a, A, neg_b, B, c_mod, C, reuse_a, reuse_b)
        c = __builtin_amdgcn_wmma_f32_16x16x4_f32(false, a, false, b,
                                                  (short)0, c, false, false);
    }

    // D layout: VGPR j -> M = j (lanes 0-15) / j+8 (lanes 16-31), N = lane&15.
    if (m < 4) {
        float* col = st + (size_t)m * N + base + half * 8;
        #pragma unroll
        for (int j = 0; j < 8; ++j) col[j] = c[j];
    }
}

// ---------------------------------------------------------------------------
// Edge kernel: one wave per edge, lane l owns features [4l, 4l+4).
// src/dst are wave-uniform -> readfirstlane so the attention-scalar loads
// and the atomic base addresses are scalar (saddr-form), minimizing VMEM
// address traffic in the dominant phase.  Gathers x[dst] once (float4/lane),
// applies both heads' diagonal W in registers, scatter-adds with
// global_atomic_add_f32.
// ---------------------------------------------------------------------------
__global__ void gat_edge(const float* __restrict__ x,
                         const float* __restrict__ st,   // (4, N)
                         const float* __restrict__ W,    // (2, F)
                         const int*   __restrict__ ei,   // (2, E)
                         long long E, int N,
                         float* __restrict__ num0, float* __restrict__ num1,
                         float* __restrict__ rs0,  float* __restrict__ rs1)
{
    const int lane = threadIdx.x & 31;
    const int wave = threadIdx.x >> 5;
    const long long e = (long long)blockIdx.x * 8 + wave;
    if (e >= E) return;

    const int src = __builtin_amdgcn_readfirstlane(ei[e]);
    const int dst = __builtin_amdgcn_readfirstlane(ei[E + e]);

    // attention weights: exp(-leaky_relu(s[src] + t[dst], 0.2))
    const float z0 = st[src] + st[(size_t)N + dst];
    const float z1 = st[(size_t)2 * N + src] + st[(size_t)3 * N + dst];
    const float w0 = __expf(-(fmaxf(z0, 0.0f) + 0.2f * fminf(z0, 0.0f)));
    const float w1 = __expf(-(fmaxf(z1, 0.0f) + 0.2f * fminf(z1, 0.0f)));

    const float4 xv = *(const float4*)(x + (size_t)dst * GAT_F + lane * 4);
    const float4 wa = *(const float4*)(W + lane * 4);            // head 0 diag
    const float4 wb = *(const float4*)(W + GAT_F + lane * 4);    // head 1 diag

    float* n0 = num0 + (size_t)src * GAT_F + lane * 4;
    float* n1 = num1 + (size_t)src * GAT_F + lane * 4;
    atomicAdd(n0 + 0, w0 * xv.x * wa.x);
    atomicAdd(n0 + 1, w0 * xv.y * wa.y);
    atomicAdd(n0 + 2, w0 * xv.z * wa.z);
    atomicAdd(n0 + 3, w0 * xv.w * wa.w);
    atomicAdd(n1 + 0, w1 * xv.x * wb.x);
    atomicAdd(n1 + 1, w1 * xv.y * wb.y);
    atomicAdd(n1 + 2, w1 * xv.z * wb.z);
    atomicAdd(n1 + 3, w1 * xv.w * wb.w);
    if (lane == 0) {
        atomicAdd(rs0 + src, w0);
        atomicAdd(rs1 + src, w1);
    }
}

// ---------------------------------------------------------------------------
// Finalize: out = 0.5*(num0/rs0 + num1/rs1), optional ELU (layer 1 only).
// ---------------------------------------------------------------------------
__global__ void gat_finalize(const float* __restrict__ num0,
                             const float* __restrict__ num1,
                             const float* __restrict__ rs0,
                             const float* __restrict__ rs1,
                             float* __restrict__ out, int N, int apply_elu)
{
    const long long t = (long long)blockIdx.x * blockDim.x + threadIdx.x;
    const long long total = (long long)N * (GAT_F / 4);
    if (t >= total) return;
    const int node = (int)(t >> 5);
    const int q    = (int)(t & 31) * 4;

    const float i0 = 0.5f / rs0[node];
    const float i1 = 0.5f / rs1[node];
    const float4 a = *(const float4*)(num0 + (size_t)node * GAT_F + q);
    const float4 b = *(const float4*)(num1 + (size_t)node * GAT_F + q);
    float4 o;
    o.x = a.x * i0 + b.x * i1;
    o.y = a.y * i0 + b.y * i1;
    o.z = a.z * i0 + b.z * i1;
    o.w = a.w * i0 + b.w * i1;
    if (apply_elu) {
        o.x = o.x > 0.0f ? o.x : __expf(o.x) - 1.0f;
        o.y = o.y > 0.0f ? o.y : __expf(o.y) - 1.0f;
        o.z = o.z > 0.0f ? o.z : __expf(o.z) - 1.0f;
        o.w = o.w > 0.0f ? o.w : __expf(o.w) - 1.0f;
    }
    *(float4*)(out + (size_t)node * GAT_F + q) = o;
}

extern "C" void kernel_launch(void* const* d_in, const int* in_sizes, int n_in,
                              void* d_out, int out_size, void* d_ws, size_t ws_size,
                              hipStream_t stream) {
    const int       F  = GAT_F;
    const int       N  = in_sizes[0] / F;
    const long long E  = (long long)in_sizes[5] / 2;

    const float* x  = (const float*)d_in[0];
    const float* W1 = (const float*)d_in[1];   // (H,1,F) contiguous -> (2,F)
    const float* A1 = (const float*)d_in[2];   // (H,2F,1) contiguous -> (2,2F)
    const float* W2 = (const float*)d_in[3];
    const float* A2 = (const float*)d_in[4];
    const int*   ei = (const int*)d_in[5];
    float* out = (float*)d_out;

    // Workspace layout (floats): x2 | num0 | num1 | rs0 | rs1 | st
    float* ws   = (float*)d_ws;
    float* x2   = ws;                            // N*F
    float* num0 = x2   + (size_t)N * F;          // N*F
    float* num1 = num0 + (size_t)N * F;          // N*F
    float* rs0  = num1 + (size_t)N * F;          // N
    float* rs1  = rs0  + N;                      // N
    float* st   = rs1  + N;                      // 4*N
    const size_t accum_bytes = ((size_t)2 * N * F + 2 * (size_t)N) * sizeof(float);

    const dim3 blk(256);
    const int nodeBlocks = (N + 127) / 128;                       // 8 waves x 16 nodes
    const int edgeBlocks = (int)((E + 7) / 8);                    // 8 waves x 1 edge
    const int finBlocks  = (int)(((long long)N * (F / 4) + 255) / 256);

    // ---- layer 1 ----
    hipMemsetAsync(num0, 0, accum_bytes, stream);
    gat_node_wmma<<<nodeBlocks, blk, 0, stream>>>(x, W1, A1, st, N);
    gat_edge<<<edgeBlocks, blk, 0, stream>>>(x, st, W1, ei, E, N, num0, num1, rs0, rs1);
    gat_finalize<<<finBlocks, blk, 0, stream>>>(num0, num1, rs0, rs1, x2, N, 1);

    // ---- layer 2 ----
    hipMemsetAsync(num0, 0, accum_bytes, stream);
    gat_node_wmma<<<nodeBlocks, blk, 0, stream>>>(x2, W2, A2, st, N);
    gat_edge<<<edgeBlocks, blk, 0, stream>>>(x2, st, W2, ei, E, N, num0, num1, rs0, rs1);
    gat_finalize<<<finBlocks, blk, 0, stream>>>(num0, num1, rs0, rs1, out, N, 0);
}